// KanGNN_80058190397885
// MI455X (gfx1250) — compile-verified
//
#include <hip/hip_runtime.h>

typedef float v2f __attribute__((ext_vector_type(2)));
typedef float v8f __attribute__((ext_vector_type(8)));
typedef unsigned int u32x4 __attribute__((ext_vector_type(4)));
typedef int i32x4 __attribute__((ext_vector_type(4)));
typedef int i32x8 __attribute__((ext_vector_type(8)));

#define N_NODES   100000
#define IN_FEAT   128
#define HID       64
#define OUT_FEAT  40
#define GRIDK     8
#define N_EDGES   1600000
#define ROWS_PER_BLOCK 128
#define NBLOCKS   ((N_NODES + ROWS_PER_BLOCK - 1) / ROWS_PER_BLOCK)  /* 782  */
#define NPAD      (NBLOCKS * ROWS_PER_BLOCK)                          /* 100096 */
#define KANK      (HID * GRIDK)                                       /* 512 */
#define KCHUNK    32                                                  /* kk per LDS chunk */

#if defined(__AMDGCN__) && __has_builtin(__builtin_amdgcn_tensor_load_to_lds)
#define USE_TDM 1
#else
#define USE_TDM 0
#endif

// fp32 WMMA: D = A(16x4) * B(4x16) + C(16x16), wave32
static __device__ __forceinline__ v8f wmma_f32(v2f a, v2f b, v8f c) {
  return __builtin_amdgcn_wmma_f32_16x16x4_f32(
      /*neg_a=*/false, a, /*neg_b=*/false, b,
      /*c_mod=*/(short)0, c, /*reuse_a=*/false, /*reuse_b=*/false);
}

#if USE_TDM
// 1-D contiguous DMA: nelem fp32 from global to LDS via the Tensor Data Mover.
// D# packing per CDNA5 ISA:
//   group0: [1:0]=count=1, [63:32]=lds_addr, [120:64]=global_addr, [127:126]=type=2
//   group1: [17:16]=data_size(2 -> 4B), tensor_dim0=nelem, tensor_dim1=1,
//           tile_dim0=nelem (<=65535), tile_dim1=1, dim0_stride=nelem
//   groups 2/3: zero (tensor rank <= 2)
static __device__ __forceinline__ void tdm_load_1d(const float* gsrc,
                                                   void* lds_dst, int nelem) {
  unsigned long long ga = (unsigned long long)(uintptr_t)gsrc;
  unsigned lds_addr = (unsigned)(uintptr_t)lds_dst;     // flat addr[31:0] == LDS offset
  u32x4 g0;
  g0.x = 1u;                                            // count=1, user descriptor
  g0.y = lds_addr;
  g0.z = (unsigned)(ga & 0xFFFFFFFFu);                  // global_addr[31:0]
  g0.w = (unsigned)((ga >> 32) & 0x01FFFFFFu) | (2u << 30);  // ga[56:32] | type=2
  i32x8 g1;
  g1[0] = 2 << 16;                                      // data_size=4B
  g1[1] = (nelem & 0xFFFF) << 16;                       // tensor_dim0[15:0] @ bits63:48
  g1[2] = ((nelem >> 16) & 0xFFFF) | (1 << 16);         // tensor_dim0[31:16] | tensor_dim1=1
  g1[3] = (nelem & 0xFFFF) << 16;                       // tile_dim0 @ bits127:112
  g1[4] = 1;                                            // tile_dim1=1, tile_dim2=0
  g1[5] = nelem;                                        // tensor_dim0_stride[31:0]
  g1[6] = 0;
  g1[7] = 0;
  i32x4 z4 = {0, 0, 0, 0};
#if defined(__clang_major__) && (__clang_major__ >= 23)
  i32x8 z8 = {0, 0, 0, 0, 0, 0, 0, 0};
  __builtin_amdgcn_tensor_load_to_lds(g0, g1, z4, z4, z8, 0);
#else
  __builtin_amdgcn_tensor_load_to_lds(g0, g1, z4, z4, 0);
#endif
}
#endif

// ---------------------------------------------------------------- utilities
__global__ void zero_kernel(float* __restrict__ p, long n) {
  long i = (long)blockIdx.x * blockDim.x + threadIdx.x;
  long stride = (long)gridDim.x * blockDim.x;
  for (; i < n; i += stride) p[i] = 0.0f;
}

// coeff [2, HID(j), HID(i), GRIDK(g)] -> ct [2, KANK(kk=i*8+g), HID(j)]
__global__ void tcoeff_kernel(const float* __restrict__ coeff, float* __restrict__ ct) {
  int tid = blockIdx.x * blockDim.x + threadIdx.x;     // 2*512*64 = 65536
  if (tid >= 2 * KANK * HID) return;
  int j  = tid & (HID - 1);
  int kk = (tid >> 6) & (KANK - 1);
  int c  = tid >> 15;
  ct[tid] = coeff[(size_t)c * HID * KANK + (size_t)j * KANK + kk];
}

// ---------------------------------------------------------------- h = x @ W_in + b
__global__ __launch_bounds__(256) void lin_in_kernel(
    const float* __restrict__ x, const float* __restrict__ Win,
    const float* __restrict__ bin, float* __restrict__ h) {
  __shared__ float sW[IN_FEAT * HID];                    // 32 KB, whole W_in
  for (int idx = threadIdx.x; idx < IN_FEAT * HID; idx += 256)
    sW[idx] = Win[idx];
  __syncthreads();

  int lane = threadIdx.x & 31, wave = threadIdx.x >> 5;
  int row0 = blockIdx.x * ROWS_PER_BLOCK + wave * 16;
  int m = lane & 15, kh = lane >> 4;
  int r = row0 + m;
  bool rv = (r < N_NODES);
  int rc = rv ? r : (N_NODES - 1);                       // clamp: load always legal
  const float* xrow = x + (size_t)rc * IN_FEAT;
  float msk = rv ? 1.0f : 0.0f;

  v8f acc[4];
  v8f z = {0, 0, 0, 0, 0, 0, 0, 0};
#pragma unroll
  for (int jt = 0; jt < 4; jt++) acc[jt] = z;

  for (int k0 = 0; k0 < IN_FEAT; k0 += 4) {
    int k = k0 + 2 * kh;                                 // this lane's K pair
    v2f a = *(const v2f*)(xrow + k);                     // single b64 load
    a.x *= msk; a.y *= msk;                              // cndmask/mul, no branch
#pragma unroll
    for (int jt = 0; jt < 4; jt++) {
      int col = jt * 16 + m;
      v2f b;
      b.x = sW[k * HID + col];
      b.y = sW[(k + 1) * HID + col];
      acc[jt] = wmma_f32(a, b, acc[jt]);
    }
  }
  // C/D layout: VGPR rr -> rows rr (lanes 0-15) and rr+8 (lanes 16-31), N = lane&15
#pragma unroll
  for (int jt = 0; jt < 4; jt++) {
    int col = jt * 16 + m;
    float bias = bin[col];
#pragma unroll
    for (int rr = 0; rr < 8; rr++) {
      int row = row0 + rr + 8 * kh;                      // row < NPAD: padded ws
      h[(size_t)row * HID + col] = acc[jt][rr] + bias;
    }
  }
}

// ---------------------------------------------------------------- out[r] += w_e * h[c]
// lane-coalesced edge metadata, 32 broadcast iterations per wave
__global__ __launch_bounds__(256) void spmm_kernel(
    const int* __restrict__ erow, const int* __restrict__ ecol,
    const float* __restrict__ ew, const float* __restrict__ h,
    float* __restrict__ out) {
  int gid = blockIdx.x * blockDim.x + threadIdx.x;       // grid covers exactly N_EDGES
  int lane = threadIdx.x & 31;
  int r = erow[gid];
  int c = ecol[gid];
  float wt = ew[gid];
#pragma unroll 4
  for (int i = 0; i < 32; i++) {
    int   ri = __shfl(r, i, 32);                         // v_readlane broadcast
    int   ci = __shfl(c, i, 32);
    float wi = __shfl(wt, i, 32);
    float v0 = h[(size_t)ci * HID + lane];
    float v1 = h[(size_t)ci * HID + 32 + lane];
    unsafeAtomicAdd(&out[(size_t)ri * HID + lane],      wi * v0);
    unsafeAtomicAdd(&out[(size_t)ri * HID + 32 + lane], wi * v1);
  }
}

// ---------------------------------------------------------------- Fourier-KAN layer as dense WMMA GEMM
// y[b,j] = sum_kk cos-feat[b,kk]*Ct0[kk,j] + sin-feat[b,kk]*Ct1[kk,j], kk = i*8+g
__global__ __launch_bounds__(256) void kan_kernel(
    const float* __restrict__ s, const float* __restrict__ ct,
    float* __restrict__ y) {
  __shared__ float sCt[2 * KCHUNK * HID];                // 16 KB: chunk of Ct0|Ct1
  __shared__ float sS[ROWS_PER_BLOCK * 65];              // 33.3 KB, pad 65 vs banks

  // stage the 128x64 s-tile (coalesced global, padded LDS stride)
  for (int idx = threadIdx.x; idx < ROWS_PER_BLOCK * HID; idx += 256) {
    int rr = idx >> 6, col = idx & 63;
    sS[rr * 65 + col] =
        s[((size_t)blockIdx.x * ROWS_PER_BLOCK + rr) * HID + col];
  }

  int lane = threadIdx.x & 31, wave = threadIdx.x >> 5;
  int row0 = blockIdx.x * ROWS_PER_BLOCK + wave * 16;
  int m = lane & 15, kh = lane >> 4;
  const float* srowL = sS + (size_t)(wave * 16 + m) * 65;

  v8f acc[4];
  v8f z = {0, 0, 0, 0, 0, 0, 0, 0};
#pragma unroll
  for (int jt = 0; jt < 4; jt++) acc[jt] = z;

  for (int c0 = 0; c0 < KANK; c0 += KCHUNK) {
    __syncthreads();                                     // prev chunk fully consumed
#if USE_TDM
    if (wave == 0) {                                     // one wave issues both DMAs
      tdm_load_1d(ct + (size_t)c0 * HID,               &sCt[0],    KCHUNK * HID);
      tdm_load_1d(ct + (size_t)(KANK + c0) * HID,      &sCt[2048], KCHUNK * HID);
      __builtin_amdgcn_s_wait_tensorcnt(0);              // TENSORcnt -> 0
    }
#else
    for (int idx = threadIdx.x; idx < 2 * KCHUNK * HID; idx += 256) {
      int comp = idx >> 11;                              // KCHUNK*HID = 2048
      int rem  = idx & 2047;
      sCt[idx] = ct[(size_t)comp * KANK * HID +
                    (size_t)(c0 + (rem >> 6)) * HID + (rem & 63)];
    }
#endif
    __syncthreads();                                     // publish chunk to all waves

    for (int k0 = c0; k0 < c0 + KCHUNK; k0 += 4) {
      int kk  = k0 + 2 * kh;                             // lane's K pair: kk, kk+1
      int kkl = kk - c0;
      // all kk in step share input feature i = k0>>3 (k0 mod 8 in {0,4})
      float sv = srowL[k0 >> 3];
      float f0 = (float)((kk & 7) + 1);                  // harmonic g+1
      v2f ac, as;
      ac.x = __cosf(sv * f0);          as.x = __sinf(sv * f0);
      ac.y = __cosf(sv * (f0 + 1.f));  as.y = __sinf(sv * (f0 + 1.f));
#pragma unroll
      for (int jt = 0; jt < 4; jt++) {
        int col = jt * 16 + m;
        v2f bc, bs;
        bc.x = sCt[kkl * HID + col];
        bc.y = sCt[(kkl + 1) * HID + col];
        bs.x = sCt[2048 + kkl * HID + col];
        bs.y = sCt[2048 + (kkl + 1) * HID + col];
        acc[jt] = wmma_f32(ac, bc, acc[jt]);
        acc[jt] = wmma_f32(as, bs, acc[jt]);
      }
    }
  }
#pragma unroll
  for (int jt = 0; jt < 4; jt++) {
    int col = jt * 16 + m;
#pragma unroll
    for (int rr = 0; rr < 8; rr++) {
      int row = row0 + rr + 8 * kh;
      y[(size_t)row * HID + col] = acc[jt][rr];
    }
  }
}

// ---------------------------------------------------------------- logits = h @ W_out (cols padded to 48)
__global__ __launch_bounds__(256) void wout_kernel(
    const float* __restrict__ h, const float* __restrict__ Wout,
    float* __restrict__ logits) {
  int lane = threadIdx.x & 31, wave = threadIdx.x >> 5;
  int row0 = blockIdx.x * ROWS_PER_BLOCK + wave * 16;
  int m = lane & 15, kh = lane >> 4;
  const float* hrow = h + (size_t)(row0 + m) * HID;      // padded ws, safe

  v8f acc[3];
  v8f z = {0, 0, 0, 0, 0, 0, 0, 0};
#pragma unroll
  for (int jt = 0; jt < 3; jt++) acc[jt] = z;

  for (int k0 = 0; k0 < HID; k0 += 4) {
    int k = k0 + 2 * kh;
    v2f a = *(const v2f*)(hrow + k);                     // single b64 load
#pragma unroll
    for (int jt = 0; jt < 3; jt++) {
      int col = jt * 16 + m;
      bool cv  = (col < OUT_FEAT);
      int colc = cv ? col : (OUT_FEAT - 1);              // clamp: load legal
      float b0 = Wout[(size_t)k * OUT_FEAT + colc];
      float b1 = Wout[(size_t)(k + 1) * OUT_FEAT + colc];
      v2f b;
      b.x = cv ? b0 : 0.0f;                              // cndmask, no branch
      b.y = cv ? b1 : 0.0f;
      acc[jt] = wmma_f32(a, b, acc[jt]);
    }
  }
#pragma unroll
  for (int jt = 0; jt < 3; jt++) {
    int col = jt * 16 + m;
#pragma unroll
    for (int rr = 0; rr < 8; rr++) {
      int row = row0 + rr + 8 * kh;
      logits[(size_t)row * 48 + col] = acc[jt][rr];
    }
  }
}

// ---------------------------------------------------------------- row-wise log_softmax over 40 classes
__global__ void lsm_kernel(const float* __restrict__ logits, float* __restrict__ out) {
  int r = blockIdx.x * blockDim.x + threadIdx.x;
  if (r >= N_NODES) return;
  const float* lp = logits + (size_t)r * 48;
  float mx = lp[0];
#pragma unroll
  for (int j = 1; j < OUT_FEAT; j++) mx = fmaxf(mx, lp[j]);
  float sum = 0.0f;
#pragma unroll
  for (int j = 0; j < OUT_FEAT; j++) sum += __expf(lp[j] - mx);
  float lse = mx + __logf(sum);
  float* op = out + (size_t)r * OUT_FEAT;
#pragma unroll
  for (int j = 0; j < OUT_FEAT; j++) op[j] = lp[j] - lse;
}

// ---------------------------------------------------------------- driver
extern "C" void kernel_launch(void* const* d_in, const int* in_sizes, int n_in,
                              void* d_out, int out_size, void* d_ws, size_t ws_size,
                              hipStream_t stream) {
  const float* x    = (const float*)d_in[0];
  const int*   erow = (const int*)d_in[1];
  const int*   ecol = (const int*)d_in[2];
  const float* ew   = (const float*)d_in[3];
  const float* Win  = (const float*)d_in[4];
  const float* bin  = (const float*)d_in[5];
  const float* c1   = (const float*)d_in[6];
  const float* c2   = (const float*)d_in[7];
  const float* Wout = (const float*)d_in[8];
  float* out = (float*)d_out;

  // workspace layout (floats): bufA | bufB | Ct1 | Ct2 | logits
  float* bufA   = (float*)d_ws;
  float* bufB   = bufA + (size_t)NPAD * HID;
  float* ct1    = bufB + (size_t)NPAD * HID;
  float* ct2    = ct1  + (size_t)2 * KANK * HID;
  float* logits = ct2  + (size_t)2 * KANK * HID;

  const long nodebuf = (long)NPAD * HID;
  const int tcBlocks = (2 * KANK * HID + 255) / 256;
  const int spBlocks = N_EDGES / 256;                    // exact: 6250

  // coeff transposes (coalesced B staging for the KAN GEMMs)
  tcoeff_kernel<<<tcBlocks, 256, 0, stream>>>(c1, ct1);
  tcoeff_kernel<<<tcBlocks, 256, 0, stream>>>(c2, ct2);

  // h = x @ W_in + b        -> bufA
  lin_in_kernel<<<NBLOCKS, 256, 0, stream>>>(x, Win, bin, bufA);

  // s1 = spmm(h)            -> bufB
  zero_kernel<<<2048, 256, 0, stream>>>(bufB, nodebuf);
  spmm_kernel<<<spBlocks, 256, 0, stream>>>(erow, ecol, ew, bufA, bufB);

  // h1 = kan(s1, C1)        -> bufA
  kan_kernel<<<NBLOCKS, 256, 0, stream>>>(bufB, ct1, bufA);

  // s2 = spmm(h1)           -> bufB
  zero_kernel<<<2048, 256, 0, stream>>>(bufB, nodebuf);
  spmm_kernel<<<spBlocks, 256, 0, stream>>>(erow, ecol, ew, bufA, bufB);

  // h2 = kan(s2, C2)        -> bufA
  kan_kernel<<<NBLOCKS, 256, 0, stream>>>(bufB, ct2, bufA);

  // logits = h2 @ W_out     -> ws, then log_softmax -> d_out
  wout_kernel<<<NBLOCKS, 256, 0, stream>>>(bufA, Wout, logits);
  lsm_kernel<<<(N_NODES + 255) / 256, 256, 0, stream>>>(logits, out);
}